// net_gat_84524956385825
// MI455X (gfx1250) — compile-verified
//
#include <hip/hip_runtime.h>

#define DIN 128
#define DHID 128
#define DOUTF 64
#define NEG_SLOPE 0.2f

typedef __attribute__((ext_vector_type(2))) float v2f;
typedef __attribute__((ext_vector_type(8))) float v8f;

// ---------------------------------------------------------------------------
// utility: fill a u32 buffer (used for zeroing and for encoded -inf init)
// ---------------------------------------------------------------------------
__global__ void fill_u32_kernel(unsigned int* p, unsigned int v, long long n) {
    long long i = (long long)blockIdx.x * blockDim.x + threadIdx.x;
    if (i < n) p[i] = v;
}

// ---------------------------------------------------------------------------
// WMMA f32 GEMM: C[M x Nout] = act(A[M x K] @ W[K x Nout] + bias)
// Row-major everywhere. Block owns one 16-row tile of C; each wave computes a
// 16x32 column slice (two 16x16 WMMA accumulators). The 16xK A panel is staged
// once per block into LDS via gfx1250 async global->LDS b128 copies
// (ASYNCcnt / s_wait_asynccnt), then read per-lane as ds_load_b64, while B
// columns stream from global (L2-resident weights).
// Requirements met by this net: M%16==0 (M=50000), Nout%32==0 (512/128/64),
// K%4==0 (512/128/64), and 16*K/4 divisible by blockDim (64..256 threads).
// ---------------------------------------------------------------------------
__global__ void wmma_gemm_kernel(const float* __restrict__ A,
                                 const float* __restrict__ W,
                                 const float* __restrict__ bias,
                                 float* __restrict__ C,
                                 int M, int K, int Nout, int act_relu) {
    extern __shared__ float smemA[];                // 16 x K floats, row-major

    const int colgrp_n = Nout >> 5;                 // 32-column groups
    const int wpb      = blockDim.x >> 5;           // waves per block (<= colgrp_n)
    const int cblocks  = colgrp_n / wpb;            // column blocks per row tile
    const int m0       = (blockIdx.x / cblocks) << 4;
    const int cb       = blockIdx.x % cblocks;
    const int waveid   = threadIdx.x >> 5;
    const int n0       = (cb * wpb + waveid) << 5;

    // ---- stage A tile (16 x K) into LDS with async global->LDS copies ----
    const int tid   = threadIdx.x;
    const int T     = blockDim.x;
    const int k4    = K >> 2;                       // float4 chunks per row
    const int nIter = (16 * k4) / T;                // exact & uniform for our shapes
    unsigned lbase = (unsigned)(unsigned long long)
        (__attribute__((address_space(3))) float*)smemA;
    for (int it = 0; it < nIter; ++it) {
        int fidx = it * T + tid;
        int row  = fidx / k4;
        int c4   = fidx - row * k4;
        unsigned laddr = lbase + (unsigned)fidx * 16u;
        unsigned long long gaddr =
            (unsigned long long)(A + (size_t)(m0 + row) * K + (size_t)c4 * 4);
        asm volatile("global_load_async_to_lds_b128 %0, %1, off"
                     :: "v"(laddr), "v"(gaddr) : "memory");
    }
    asm volatile("s_wait_asynccnt 0x0" ::: "memory");
    __syncthreads();

    // ---- 16x32 WMMA tile per wave ----
    const int lane = threadIdx.x & 31;
    const int half = lane >> 4;                     // 0: K pair 0/1, 1: K pair 2/3
    const int l15  = lane & 15;

    const float* aptr  = smemA + (size_t)l15 * K + 2 * half;      // ds_load_b64
    const float* bcol0 = W + (size_t)(2 * half) * Nout + (n0 + l15);
    const float* bcol1 = bcol0 + 16;

    v8f acc0 = {0.f, 0.f, 0.f, 0.f, 0.f, 0.f, 0.f, 0.f};
    v8f acc1 = {0.f, 0.f, 0.f, 0.f, 0.f, 0.f, 0.f, 0.f};

    for (int k = 0; k < K; k += 4) {
        float2 av = *(const float2*)(aptr + k);
        v2f a  = { av.x, av.y };
        v2f b0 = { bcol0[0], bcol0[Nout] };
        v2f b1 = { bcol1[0], bcol1[Nout] };
        acc0 = __builtin_amdgcn_wmma_f32_16x16x4_f32(false, a, false, b0,
                                                     (short)0, acc0, false, false);
        acc1 = __builtin_amdgcn_wmma_f32_16x16x4_f32(false, a, false, b1,
                                                     (short)0, acc1, false, false);
        bcol0 += (size_t)4 * Nout;
        bcol1 += (size_t)4 * Nout;
    }

    const int   col0 = n0 + l15;
    const float bb0  = bias ? bias[col0]      : 0.f;
    const float bb1  = bias ? bias[col0 + 16] : 0.f;
    float* cbase = C + (size_t)(m0 + half * 8) * Nout + col0;
#pragma unroll
    for (int r = 0; r < 8; ++r) {
        float v0 = acc0[r] + bb0;
        float v1 = acc1[r] + bb1;
        if (act_relu) { v0 = fmaxf(v0, 0.f); v1 = fmaxf(v1, 0.f); }
        cbase[(size_t)r * Nout]      = v0;
        cbase[(size_t)r * Nout + 16] = v1;
    }
}

// ---------------------------------------------------------------------------
// parsing = relu(SCALE * parsing0)
// ---------------------------------------------------------------------------
__global__ void parsing_kernel(const float* __restrict__ p0, float* __restrict__ p, int n) {
    int i = blockIdx.x * blockDim.x + threadIdx.x;
    if (i < n) p[i] = fmaxf(2.0f * p0[i], 0.f);
}

// ---------------------------------------------------------------------------
// per-node attention dots: a_s[i] = h[i]·a_src, a_d[i] = h[i]·a_dst
// ---------------------------------------------------------------------------
__global__ void node_dots_kernel(const float* __restrict__ h,
                                 const float* __restrict__ asrc,
                                 const float* __restrict__ adst,
                                 float* __restrict__ a_s, float* __restrict__ a_d,
                                 int n, int F) {
    int i = blockIdx.x * blockDim.x + threadIdx.x;
    if (i >= n) return;
    const float* hr = h + (size_t)i * F;
    float s0 = 0.f, s1 = 0.f;
    for (int f = 0; f < F; ++f) {
        float hv = hr[f];
        s0 += hv * asrc[f];
        s1 += hv * adst[f];
    }
    a_s[i] = s0;
    a_d[i] = s1;
}

// ---------------------------------------------------------------------------
// edge gate raw value + global sum/sumsq via block reduction
// ew_raw[e] = dot(logits[src], g[dst]);  g = logits @ parsing (precomputed)
// ---------------------------------------------------------------------------
__global__ void edge_ew_kernel(const int* __restrict__ src, const int* __restrict__ dst,
                               const float* __restrict__ logits, const float* __restrict__ g,
                               float* __restrict__ ew_raw, float* __restrict__ sums, int E) {
    int e = blockIdx.x * blockDim.x + threadIdx.x;
    float v = 0.f;
    if (e < E) {
        const float* ls = logits + (size_t)src[e] * DOUTF;
        const float* gd = g      + (size_t)dst[e] * DOUTF;
        for (int c = 0; c < DOUTF; c += 4) {
            v += ls[c]   * gd[c]   + ls[c+1] * gd[c+1]
               + ls[c+2] * gd[c+2] + ls[c+3] * gd[c+3];
        }
        ew_raw[e] = v;
    }
    __shared__ float red0[256];
    __shared__ float red1[256];
    red0[threadIdx.x] = (e < E) ? v     : 0.f;
    red1[threadIdx.x] = (e < E) ? v * v : 0.f;
    __syncthreads();
    for (int s = 128; s > 0; s >>= 1) {
        if ((int)threadIdx.x < s) {
            red0[threadIdx.x] += red0[threadIdx.x + s];
            red1[threadIdx.x] += red1[threadIdx.x + s];
        }
        __syncthreads();
    }
    if (threadIdx.x == 0) {
        atomicAdd(&sums[0], red0[0]);
        atomicAdd(&sums[1], red1[0]);
    }
}

// ew = (raw - mean) * sqrt(1e-4 / var_unbiased) + 1
__global__ void edge_norm_kernel(float* __restrict__ ew, const float* __restrict__ sums, int E) {
    int e = blockIdx.x * blockDim.x + threadIdx.x;
    if (e >= E) return;
    float mean = sums[0] / (float)E;
    float var  = (sums[1] - sums[0] * mean) / (float)(E - 1);
    float sc   = sqrtf(1e-4f / var);
    ew[e] = (ew[e] - mean) * sc + 1.0f;
}

// ---------------------------------------------------------------------------
// ordered-uint encoding for float atomic max (handles negatives correctly)
// ---------------------------------------------------------------------------
__device__ __forceinline__ unsigned int fenc(float v) {
    unsigned int b = __float_as_uint(v);
    return (b & 0x80000000u) ? ~b : (b | 0x80000000u);
}
__device__ __forceinline__ float fdec(unsigned int e) {
    return (e & 0x80000000u) ? __uint_as_float(e ^ 0x80000000u) : __uint_as_float(~e);
}
#define FENC_NEG_INF 0x007FFFFFu  // fenc(-inf)

// alpha = leaky_relu(a_s[src]+a_d[dst]); segment max over dst
__global__ void edge_alpha_kernel(const int* __restrict__ src, const int* __restrict__ dst,
                                  const float* __restrict__ a_s, const float* __restrict__ a_d,
                                  float* __restrict__ alpha, unsigned int* __restrict__ m_enc,
                                  int E) {
    int e = blockIdx.x * blockDim.x + threadIdx.x;
    if (e >= E) return;
    float v = a_s[src[e]] + a_d[dst[e]];
    v = (v >= 0.f) ? v : NEG_SLOPE * v;
    alpha[e] = v;
    atomicMax(&m_enc[dst[e]], fenc(v));
}

// e = exp(alpha - m[dst]) (in place over alpha); segment sum over dst
__global__ void edge_exp_kernel(const int* __restrict__ dst, float* __restrict__ alpha_e,
                                const unsigned int* __restrict__ m_enc,
                                float* __restrict__ s_sum, int E) {
    int e = blockIdx.x * blockDim.x + threadIdx.x;
    if (e >= E) return;
    int d = dst[e];
    float ex = __expf(alpha_e[e] - fdec(m_enc[d]));
    alpha_e[e] = ex;
    atomicAdd(&s_sum[d], ex);
}

// out[n, :] = bias broadcast (accumulator init)
__global__ void bias_init_kernel(float* __restrict__ out, const float* __restrict__ bias,
                                 long long total, int F) {
    long long i = (long long)blockIdx.x * blockDim.x + threadIdx.x;
    if (i >= total) return;
    out[i] = bias[(int)(i % F)];
}

// out[dst] += (e/(s[dst]+1e-16)) * ew * h[src]  (float4 per thread)
__global__ void edge_scatter_kernel(const int* __restrict__ src, const int* __restrict__ dst,
                                    const float* __restrict__ e_exp, const float* __restrict__ s_sum,
                                    const float* __restrict__ ew, const float* __restrict__ h,
                                    float* __restrict__ out, int E, int F) {
    const int groups = F >> 2;
    long long t = (long long)blockIdx.x * blockDim.x + threadIdx.x;
    if (t >= (long long)E * groups) return;
    int e    = (int)(t / groups);
    int gofs = (int)(t % groups) << 2;
    int sN = src[e], dN = dst[e];
    float attn = e_exp[e] / (s_sum[dN] + 1e-16f) * ew[e];
    const float4 hv = *(const float4*)(h + (size_t)sN * F + gofs);
    float* o = out + (size_t)dN * F + gofs;
    atomicAdd(o + 0, attn * hv.x);
    atomicAdd(o + 1, attn * hv.y);
    atomicAdd(o + 2, attn * hv.z);
    atomicAdd(o + 3, attn * hv.w);
}

// elementwise epilogues
__global__ void relu_kernel(const float* __restrict__ x, float* __restrict__ y, long long n) {
    long long i = (long long)blockIdx.x * blockDim.x + threadIdx.x;
    if (i < n) y[i] = fmaxf(x[i], 0.f);
}
__global__ void relu_add_kernel(const float* __restrict__ x, const float* __restrict__ res,
                                float* __restrict__ y, long long n) {
    long long i = (long long)blockIdx.x * blockDim.x + threadIdx.x;
    if (i < n) y[i] = fmaxf(x[i], 0.f) + res[i];
}

// ---------------------------------------------------------------------------
// host driver
// ---------------------------------------------------------------------------
extern "C" void kernel_launch(void* const* d_in, const int* in_sizes, int n_in,
                              void* d_out, int out_size, void* d_ws, size_t ws_size,
                              hipStream_t stream) {
    const int N = in_sizes[0] / DIN;   // 50000
    const int E = in_sizes[1] / 2;     // 800000

    const float* x    = (const float*)d_in[0];
    const int*   ei   = (const int*)d_in[1];
    const int*   src  = ei;
    const int*   dst  = ei + E;
    const float* W_m1 = (const float*)d_in[2];
    const float* b_m1 = (const float*)d_in[3];
    const float* W_m2 = (const float*)d_in[4];
    const float* b_m2 = (const float*)d_in[5];
    const float* W_m3 = (const float*)d_in[6];
    const float* b_m3 = (const float*)d_in[7];
    const float* pars0= (const float*)d_in[8];
    const float* W0   = (const float*)d_in[9];
    const float* as0  = (const float*)d_in[10];
    const float* ad0  = (const float*)d_in[11];
    const float* b0   = (const float*)d_in[12];
    const float* W1   = (const float*)d_in[13];
    const float* as1  = (const float*)d_in[14];
    const float* ad1  = (const float*)d_in[15];
    const float* b1   = (const float*)d_in[16];
    const float* W2   = (const float*)d_in[17];
    const float* as2  = (const float*)d_in[18];
    const float* ad2  = (const float*)d_in[19];
    const float* b2   = (const float*)d_in[20];

    // workspace layout (floats), with reuse
    float* ws = (float*)d_ws;
    size_t o = 0;
    float* R0      = ws + o; o += (size_t)N * 512;  // h1; later logits(+g); later h + out1
    float* h2      = ws + o; o += (size_t)N * 64;
    float* ew      = ws + o; o += (size_t)E;
    float* edgetmp = ws + o; o += (size_t)E;        // alpha, then exp(alpha-m)
    float* a_s     = ws + o; o += (size_t)N;
    float* a_d     = ws + o; o += (size_t)N;
    float* m_enc   = ws + o; o += (size_t)N;        // encoded segment max
    float* s_sum   = ws + o; o += (size_t)N;
    float* x0      = ws + o; o += (size_t)N * 128;  // layer0 output (pre-relu, residual)
    float* xcur    = ws + o; o += (size_t)N * 128;  // activated layer inputs
    float* parsing = ws + o; o += 64 * 64;
    float* sums    = ws + o; o += 8;
    (void)ws_size; (void)n_in; (void)out_size;

    float* logits = R0;                       // after h1 is dead
    float* g      = R0 + (size_t)N * 64;
    float* hbuf   = R0;                       // per-layer projected features
    float* out1   = R0 + (size_t)N * 128;     // layer1 output

    const int TB = 256;
    auto gemm = [&](const float* A, const float* Wm, const float* b, float* Cout,
                    int M, int K, int Nout, int relu) {
        int colgrp  = Nout >> 5;
        int wpb     = colgrp < 8 ? colgrp : 8;      // waves per block
        int cblocks = colgrp / wpb;
        int blocks  = (M >> 4) * cblocks;
        size_t shmem = (size_t)16 * K * sizeof(float);  // A panel
        wmma_gemm_kernel<<<blocks, wpb * 32, shmem, stream>>>(A, Wm, b, Cout,
                                                              M, K, Nout, relu);
    };
    auto fillu = [&](float* p, unsigned int v, long long n) {
        fill_u32_kernel<<<(int)((n + TB - 1) / TB), TB, 0, stream>>>((unsigned int*)p, v, n);
    };

    // ---- learn_soft_edge: MLP + bilinear edge gate ----
    gemm(x,      W_m1, b_m1, R0,     N, DIN, 512, 1);    // h1 = relu(x@W_m1+b)
    gemm(R0,     W_m2, b_m2, h2,     N, 512, 64,  1);    // h2 = relu(h1@W_m2+b)
    gemm(h2,     W_m3, b_m3, logits, N, 64,  64,  0);    // logits (overwrites dead h1)
    parsing_kernel<<<(64 * 64 + TB - 1) / TB, TB, 0, stream>>>(pars0, parsing, 64 * 64);
    gemm(logits, parsing, nullptr, g, N, 64,  64,  0);   // g = logits @ parsing

    fillu(sums, 0u, 2);
    edge_ew_kernel<<<(E + TB - 1) / TB, TB, 0, stream>>>(src, dst, logits, g, ew, sums, E);
    edge_norm_kernel<<<(E + TB - 1) / TB, TB, 0, stream>>>(ew, sums, E);

    // ---- one GAT layer ----
    auto gat = [&](const float* xin, const float* W, const float* aw_s, const float* aw_d,
                   const float* bias, int Fin, int Fout, float* outbuf) {
        gemm(xin, W, nullptr, hbuf, N, Fin, Fout, 0);
        node_dots_kernel<<<(N + TB - 1) / TB, TB, 0, stream>>>(hbuf, aw_s, aw_d, a_s, a_d, N, Fout);
        fillu(m_enc, FENC_NEG_INF, N);
        fillu(s_sum, 0u, N);
        edge_alpha_kernel<<<(E + TB - 1) / TB, TB, 0, stream>>>(src, dst, a_s, a_d,
                                                                edgetmp, (unsigned int*)m_enc, E);
        edge_exp_kernel<<<(E + TB - 1) / TB, TB, 0, stream>>>(dst, edgetmp,
                                                              (unsigned int*)m_enc, s_sum, E);
        long long tot = (long long)N * Fout;
        bias_init_kernel<<<(int)((tot + TB - 1) / TB), TB, 0, stream>>>(outbuf, bias, tot, Fout);
        long long st = (long long)E * (Fout >> 2);
        edge_scatter_kernel<<<(int)((st + TB - 1) / TB), TB, 0, stream>>>(src, dst, edgetmp, s_sum,
                                                                          ew, hbuf, outbuf, E, Fout);
    };

    // layer 0: x -> x0 (keep pre-relu as residual)
    gat(x, W0, as0, ad0, b0, DIN, DHID, x0);
    relu_kernel<<<(int)(((long long)N * DHID + TB - 1) / TB), TB, 0, stream>>>(x0, xcur, (long long)N * DHID);
    // layer 1: relu(x0) -> out1 ; x1 = relu(out1) + x0
    gat(xcur, W1, as1, ad1, b1, DHID, DHID, out1);
    relu_add_kernel<<<(int)(((long long)N * DHID + TB - 1) / TB), TB, 0, stream>>>(out1, x0, xcur, (long long)N * DHID);
    // layer 2: -> d_out
    gat(xcur, W2, as2, ad2, b2, DHID, DOUTF, (float*)d_out);
}